// CFAR_OSCA_2D_old_75849122448293
// MI455X (gfx1250) — compile-verified
//
#include <hip/hip_runtime.h>
#include <hip/hip_bf16.h>
#include <math.h>

// ---------------------------------------------------------------------------
// CFAR OS (range) + CA (velocity) for data (B=16, V=256, R=1024) float32.
//
// Pass 1 (os_cfar_kernel): per (b,v) row, circularly pad by HR=20, compute
//   sorted top-8 of every contiguous 16-run S(p) once (shared by 2 cells),
//   then miu = 8th largest of S(r) ∪ S(r+25) via sorted-merge closed form.
//   os = OS_ALPHA * miu  ->  d_ws.
// Pass 2 (ca_wmma_kernel): 16-tap circular CA sum over velocity expressed as
//   a banded 16x36 (0/1) x 36x16 matmul, executed as 9 chained
//   v_wmma_f32_16x16x4_f32 per 16x16 output tile, B operands staged in LDS.
// ---------------------------------------------------------------------------

#define BB 16
#define VV 256
#define RR 1024
#define HRP 20                    // range halo = G[1]+T[1]
#define PADLEN (RR + 2 * HRP)     // 1064
#define NS (RR + 25)              // 1049 contiguous-16 run starts per row
#define S_STRIDE 9                // pad sorted-8 rows to 9 floats (banking)
#define OS_ALPHA 3.367842f        // sqrt of OS-CFAR threshold (PFA=1e-5, k=24, n=32)

typedef __attribute__((ext_vector_type(2))) float v2f;
typedef __attribute__((ext_vector_type(8))) float v8f;

// ------------------------------- Pass 1 ------------------------------------
__global__ __launch_bounds__(256) void os_cfar_kernel(const float* __restrict__ data,
                                                      float* __restrict__ os) {
    __shared__ float pad[PADLEN];
    __shared__ float s8[NS * S_STRIDE];

    const int tid = threadIdx.x;
    const int row = blockIdx.x;                 // b * VV + v
    const float* __restrict__ src = data + (size_t)row * RR;

    // Circularly padded row: pad[i] = data[row][(i - HR) mod R]
    for (int i = tid; i < PADLEN; i += 256) {
        pad[i] = src[(i + (RR - HRP)) & (RR - 1)];
    }
    __syncthreads();

    // Stage 1: sorted (descending) top-8 of each contiguous 16-run.
    for (int p = tid; p < NS; p += 256) {
        float m[8];
#pragma unroll
        for (int k = 0; k < 8; k++) m[k] = -3.4e38f;
#pragma unroll
        for (int j = 0; j < 16; j++) {
            float x = pad[p + j];
#pragma unroll
            for (int k = 0; k < 8; k++) {       // branchless sorted insert
                float hi = fmaxf(m[k], x);
                x = fminf(m[k], x);
                m[k] = hi;
            }
        }
#pragma unroll
        for (int k = 0; k < 8; k++) s8[p * S_STRIDE + k] = m[k];
    }
    __syncthreads();

    // Stage 2: miu = 8th largest of leading-16 (start r) ∪ trailing-16 (start r+25).
    float* __restrict__ dst = os + (size_t)row * RR;
    for (int r = tid; r < RR; r += 256) {
        const float* __restrict__ A = &s8[r * S_STRIDE];
        const float* __restrict__ Bt = &s8[(r + 25) * S_STRIDE];
        float miu = Bt[7];
#pragma unroll
        for (int i = 1; i < 8; i++) miu = fmaxf(miu, fminf(A[i - 1], Bt[7 - i]));
        miu = fmaxf(miu, A[7]);
        dst[r] = OS_ALPHA * miu;
    }
}

// ------------------------------- Pass 2 ------------------------------------
// Banded CA sum via V_WMMA_F32_16X16X4_F32.
//   out[v0+i][r0+n] = (1/16) * sum_j S[i][j] * os[(v0-10+j) mod V][r0+n]
//   S[i][j] = 1 iff (j-i) in [0,7] ∪ [13,20],  j = 0..35.
#define NJ 36
#define LDSTR 272   // 256 + 16 floats: rows 2 apart differ by 32 LDS banks

__global__ __launch_bounds__(256) void ca_wmma_kernel(const float* __restrict__ os,
                                                      float* __restrict__ out) {
    __shared__ float tile[NJ * LDSTR];

    const int tid = threadIdx.x;
    int blk = blockIdx.x;                 // blk = ((b*16 + vt)*4 + rc)
    const int rc = blk & 3;  blk >>= 2;
    const int vt = blk & 15; blk >>= 4;
    const int b  = blk;
    const int v0 = vt * 16;
    const int rbase = rc * 256;

    // Stage the 36-row velocity slab (circular in v) for this 256-range chunk.
    for (int t = tid; t < NJ * 256; t += 256) {
        const int j = t >> 8;
        const int c = t & 255;
        const int vsrc = (v0 + j + (VV - 10)) & (VV - 1);
        tile[j * LDSTR + c] = os[((size_t)(b * VV + vsrc)) * RR + rbase + c];
    }
    __syncthreads();

    const int lane = tid & 31;
    const int wave = tid >> 5;
    const int Mn = lane & 15;                  // M for A/D rows, N for B/D cols
    const int k0 = (lane < 16) ? 0 : 2;        // K pair held by this half-wave
    const int mbase = (lane < 16) ? 0 : 8;     // D rows in this half-wave

#pragma unroll
    for (int half = 0; half < 2; half++) {
        const int r0 = (wave + half * 8) * 16; // this wave's 16-wide range tile
        v8f acc = {0.f, 0.f, 0.f, 0.f, 0.f, 0.f, 0.f, 0.f};

#pragma unroll
        for (int s = 0; s < 9; s++) {          // K = 36 in steps of 4
            const int j0 = 4 * s + k0;
            const int d0 = j0 - Mn;
            const int d1 = j0 + 1 - Mn;
            v2f a;                             // banded 0/1 A fragment (registers)
            a.x = ((d0 >= 0 && d0 <= 7) || (d0 >= 13 && d0 <= 20)) ? 1.0f : 0.0f;
            a.y = ((d1 >= 0 && d1 <= 7) || (d1 >= 13 && d1 <= 20)) ? 1.0f : 0.0f;
            v2f bv;                            // B fragment from LDS slab
            bv.x = tile[j0 * LDSTR + r0 + Mn];
            bv.y = tile[(j0 + 1) * LDSTR + r0 + Mn];
            acc = __builtin_amdgcn_wmma_f32_16x16x4_f32(
                false, a, false, bv, (short)0, acc, false, false);
        }

        float* __restrict__ dst = out + (size_t)b * VV * RR + rbase + r0 + Mn;
#pragma unroll
        for (int g = 0; g < 8; g++) {          // D layout: VGPR g -> M = mbase+g
            dst[(size_t)(v0 + mbase + g) * RR] = acc[g] * 0.0625f;  // / CA_N
        }
    }
}

// ------------------------------- Launch ------------------------------------
extern "C" void kernel_launch(void* const* d_in, const int* in_sizes, int n_in,
                              void* d_out, int out_size, void* d_ws, size_t ws_size,
                              hipStream_t stream) {
    const float* data = (const float*)d_in[0];
    float* out = (float*)d_out;
    float* osbuf = (float*)d_ws;   // needs B*V*R*4 = 16 MB scratch

    os_cfar_kernel<<<dim3(BB * VV), dim3(256), 0, stream>>>(data, osbuf);
    ca_wmma_kernel<<<dim3(BB * 16 * 4), dim3(256), 0, stream>>>(osbuf, out);
}